// BinLoss_7103875908252
// MI455X (gfx1250) — compile-verified
//
#include <hip/hip_runtime.h>

typedef float v2f __attribute__((ext_vector_type(2)));
typedef float v8f __attribute__((ext_vector_type(8)));

#define N_ROWS   32768
#define DIMS     512
#define TILE_R   16
#define CHUNK    64          // columns staged per LDS pass
#define LSTRIDE  68          // padded LDS row stride (floats) -> conflict-free
#define WAVES_PB 4
#define BLOCK_T  (WAVES_PB * 32)
#define NTILES   (N_ROWS / TILE_R)      // 2048
#define NBLOCKS  (NTILES / WAVES_PB)    // 512

// searchsorted(linspace(0,1,31), v, 'left') with reference clamps.
__device__ __forceinline__ int bin32(float v) {
    int b = 0;
#pragma unroll
    for (int j = 0; j < 31; ++j)
        b += ((float)j * (1.0f / 30.0f) < v) ? 1 : 0;
    if (v > 1.0f) b = 31;
    if (v < 0.0f) b = 0;
    return b;
}

__global__ __launch_bounds__(BLOCK_T)
void binloss_main(const float* __restrict__ features,
                  const float* __restrict__ target,
                  const float* __restrict__ centers,
                  float* __restrict__ partial) {
    __shared__ float lds[WAVES_PB * 2 * TILE_R * LSTRIDE];  // 34,816 B

    const int lane = threadIdx.x & 31;
    const int wv   = threadIdx.x >> 5;
    const int tile = blockIdx.x * WAVES_PB + wv;
    const int rowBase = tile * TILE_R;

    float* ldsF = &lds[wv * (2 * TILE_R * LSTRIDE)];
    float* ldsC = ldsF + TILE_R * LSTRIDE;

    // --- labels for this tile's 16 rows (lanes 16-31 duplicate, no divergence)
    const int myRow = rowBase + (lane & 15);
    const float t0 = target[myRow * 2 + 0];
    const float t1 = target[myRow * 2 + 1];
    const int label = bin32(t0) * 32 + bin32(t1);

    v8f acc = {};
    // WMMA operand position for this lane: row M=N=lane%16, K pair at k0+2*(lane/16)
    const int mrow = lane & 15;
    const int koff = (lane >> 4) * 2;

    for (int chunk = 0; chunk < DIMS / CHUNK; ++chunk) {
        const int colBase = chunk * CHUNK;

        // --- stage: coalesced float2 loads (wave covers 64 cols/row)
#pragma unroll
        for (int r = 0; r < TILE_R; ++r) {
            const int lab = __shfl(label, r, 32);
            const float2 f2 = *(const float2*)(features +
                (size_t)(rowBase + r) * DIMS + colBase + lane * 2);
            const float2 c2 = *(const float2*)(centers +
                (size_t)lab * DIMS + colBase + lane * 2);
            *(float2*)(ldsF + r * LSTRIDE + lane * 2) = f2;
            *(float2*)(ldsC + r * LSTRIDE + lane * 2) = c2;
        }

        // --- compute: diag of (F-C)(F-C)^T via f32 WMMA; same v2f feeds A and B
#pragma unroll
        for (int k0 = 0; k0 < CHUNK; k0 += 4) {
            const float2 fa = *(const float2*)(ldsF + mrow * LSTRIDE + k0 + koff);
            const float2 ca = *(const float2*)(ldsC + mrow * LSTRIDE + k0 + koff);
            v2f d;
            d.x = fa.x - ca.x;
            d.y = fa.y - ca.y;
            acc = __builtin_amdgcn_wmma_f32_16x16x4_f32(
                /*neg_a=*/false, d, /*neg_b=*/false, d,
                /*c_mod=*/(short)0, acc, /*reuse_a=*/false, /*reuse_b=*/false);
        }
    }

    // --- extract diagonal: rows 0-7 at (vgpr j, lane j); rows 8-15 at (vgpr j, lane j+24)
    const int idx = (lane < 8) ? lane : ((lane >= 24) ? (lane - 24) : -1);
    float dv = 0.0f;
#pragma unroll
    for (int j = 0; j < 8; ++j)
        if (idx == j) dv = acc[j];

    const float sq  = fminf(fmaxf(dv, 1e-12f), 1e12f);
    float val = (idx >= 0) ? sq : 0.0f;

#pragma unroll
    for (int off = 16; off >= 1; off >>= 1)
        val += __shfl_xor(val, off, 32);

    if (lane == 0) partial[tile] = val;   // one tile-sum per tile, written every call
}

__global__ __launch_bounds__(256)
void binloss_reduce(const float* __restrict__ partial, float* __restrict__ out,
                    int n, float invN) {
    __shared__ float sm[256];
    float s = 0.0f;
    for (int i = threadIdx.x; i < n; i += 256) s += partial[i];
    sm[threadIdx.x] = s;
    __syncthreads();
#pragma unroll
    for (int st = 128; st > 0; st >>= 1) {
        if (threadIdx.x < st) sm[threadIdx.x] += sm[threadIdx.x + st];
        __syncthreads();
    }
    if (threadIdx.x == 0) out[0] = sm[0] * invN;
}

extern "C" void kernel_launch(void* const* d_in, const int* in_sizes, int n_in,
                              void* d_out, int out_size, void* d_ws, size_t ws_size,
                              hipStream_t stream) {
    (void)in_sizes; (void)n_in; (void)out_size; (void)ws_size;
    const float* features = (const float*)d_in[0];   // [32768, 512] f32
    const float* target   = (const float*)d_in[1];   // [32768, 2]   f32
    const float* centers  = (const float*)d_in[2];   // [1024, 512]  f32
    float* out     = (float*)d_out;                  // scalar f32 mean
    float* partial = (float*)d_ws;                   // 2048 f32 tile sums

    binloss_main<<<NBLOCKS, BLOCK_T, 0, stream>>>(features, target, centers, partial);
    binloss_reduce<<<1, 256, 0, stream>>>(partial, out, NTILES, 1.0f / (float)N_ROWS);
}